// GraphProcessingStream_64939905515825
// MI455X (gfx1250) — compile-verified
//
#include <hip/hip_runtime.h>
#include <hip/hip_bf16.h>
#include <math.h>

// Problem constants (from reference)
#define NNODES 100000
#define NEDGES 1600000
#define ETOT   (NNODES + NEDGES)   // edges + self loops

typedef __attribute__((ext_vector_type(2))) float v2f;
typedef __attribute__((ext_vector_type(8))) float v8f;

// ---- monotone uint key for float max-atomics (total order, incl. negatives) ----
__device__ __forceinline__ unsigned fkey(float f) {
  unsigned u = __float_as_uint(f);
  return (u & 0x80000000u) ? ~u : (u | 0x80000000u);
}
__device__ __forceinline__ float funkey(unsigned k) {
  unsigned u = (k & 0x80000000u) ? (k ^ 0x80000000u) : ~k;
  return __uint_as_float(u);
}
__device__ __forceinline__ float lrelu(float x) { return x > 0.0f ? x : 0.2f * x; }

// ---------------------------------------------------------------------------
// Layer 1 node prep: feat1[n,f] = x[n]*W1[f]; per-head attention dots;
// init emax keys / denom; zero layer-1 aggregation buffer.
// ---------------------------------------------------------------------------
__global__ void l1_node_prep(const float* __restrict__ x, const float* __restrict__ W1,
                             const float* __restrict__ asrc, const float* __restrict__ adst,
                             float* __restrict__ feat, float* __restrict__ agg,
                             float* __restrict__ as_, float* __restrict__ ad_,
                             unsigned* __restrict__ emax, float* __restrict__ denom) {
  int n = blockIdx.x * blockDim.x + threadIdx.x;
  if (n >= NNODES) return;
  float xv = x[n];
  float s[4] = {0.f, 0.f, 0.f, 0.f}, d[4] = {0.f, 0.f, 0.f, 0.f};
#pragma unroll
  for (int f = 0; f < 64; ++f) {
    float hv = xv * W1[f];
    feat[(size_t)n * 64 + f] = hv;
    agg[(size_t)n * 64 + f]  = 0.0f;
    int h = f >> 4, c = f & 15;
    s[h] += hv * asrc[h * 16 + c];
    d[h] += hv * adst[h * 16 + c];
  }
#pragma unroll
  for (int h = 0; h < 4; ++h) {
    as_[n * 4 + h]  = s[h];
    ad_[n * 4 + h]  = d[h];
    emax[n * 4 + h] = 0u;      // smallest key == identity for unsigned max
    denom[n * 4 + h] = 0.0f;
  }
}

// ---------------------------------------------------------------------------
// Edge pass A: segment max of leaky-relu logits (one thread per edge*head)
// ---------------------------------------------------------------------------
__device__ __forceinline__ void edge_endpoints(const long long* ei, int e, int& s, int& d) {
  if (e < NEDGES) { s = (int)ei[e]; d = (int)ei[NEDGES + e]; }
  else            { s = d = e - NEDGES; }                     // self loop
}

__global__ void edge_max(const long long* __restrict__ ei,
                         const float* __restrict__ as_, const float* __restrict__ ad_,
                         unsigned* __restrict__ emax) {
  long long id = (long long)blockIdx.x * blockDim.x + threadIdx.x;
  if (id >= (long long)ETOT * 4) return;
  int e = (int)(id >> 2), h = (int)(id & 3);
  int s, d; edge_endpoints(ei, e, s, d);
  float v = lrelu(as_[s * 4 + h] + ad_[d * 4 + h]);
  atomicMax(&emax[d * 4 + h], fkey(v));
}

// ---------------------------------------------------------------------------
// Edge pass B: softmax denominator
// ---------------------------------------------------------------------------
__global__ void edge_sum(const long long* __restrict__ ei,
                         const float* __restrict__ as_, const float* __restrict__ ad_,
                         const unsigned* __restrict__ emax, float* __restrict__ denom) {
  long long id = (long long)blockIdx.x * blockDim.x + threadIdx.x;
  if (id >= (long long)ETOT * 4) return;
  int e = (int)(id >> 2), h = (int)(id & 3);
  int s, d; edge_endpoints(ei, e, s, d);
  float v = lrelu(as_[s * 4 + h] + ad_[d * 4 + h]);
  atomicAdd(&denom[d * 4 + h], __expf(v - funkey(emax[d * 4 + h])));
}

// ---------------------------------------------------------------------------
// Edge pass C (heavy): scatter alpha-weighted source features.
// block = one edge, thread = one feature channel (F = H*C = blockDim.x).
// ---------------------------------------------------------------------------
template <int F, int C>
__global__ void edge_scatter(const long long* __restrict__ ei,
                             const float* __restrict__ as_, const float* __restrict__ ad_,
                             const unsigned* __restrict__ emax, const float* __restrict__ denom,
                             const float* __restrict__ feat, float* __restrict__ agg) {
  int e = blockIdx.x;
  int f = threadIdx.x;
  int s, d; edge_endpoints(ei, e, s, d);
  const float* srow = &feat[(size_t)s * F + f];
  __builtin_prefetch(srow, 0, 3);                 // global_prefetch_b8 the hot gather
  int h = f / C;
  float lv    = lrelu(as_[s * 4 + h] + ad_[d * 4 + h]);
  float alpha = __expf(lv - funkey(emax[d * 4 + h])) / denom[d * 4 + h];
  atomicAdd(&agg[(size_t)d * F + f], (*srow) * alpha);
}

// ---------------------------------------------------------------------------
// bias + tanh, in place over [N,F]
// ---------------------------------------------------------------------------
template <int F>
__global__ void bias_tanh(float* __restrict__ z, const float* __restrict__ b) {
  long long id = (long long)blockIdx.x * blockDim.x + threadIdx.x;
  if (id >= (long long)NNODES * F) return;
  z[id] = tanhf(z[id] + b[id & (F - 1)]);
}

// ---------------------------------------------------------------------------
// Layer-2 GEMM: feat2[N,128] = z1[N,64] @ W2[64,128], fp32-exact via
// V_WMMA_F32_16X16X4_F32. One wave per 16x16 output tile; 8 waves/block cover
// the 8 N-tiles of one 16-row M-tile. N = 6250*16 exactly -> no bounds checks.
// A layout (ISA 7.12.2, 32-bit 16x4): lane<16 holds K={k0,k0+1}, lane>=16 K={k0+2,k0+3}.
// D layout: VGPR v -> row v (lanes 0-15) / row v+8 (lanes 16-31), col = lane&15.
// ---------------------------------------------------------------------------
__global__ __launch_bounds__(256) void gemm_l2_wmma(const float* __restrict__ A,
                                                    const float* __restrict__ B,
                                                    float* __restrict__ D) {
  const int lane = threadIdx.x & 31;
  const int wave = threadIdx.x >> 5;      // N-tile 0..7
  const int mt   = blockIdx.x;            // M-tile 0..6249
  const int hi   = lane >> 4;             // half-wave select
  const int lo   = lane & 15;
  const int row  = mt * 16 + lo;          // A row for this lane
  const int col  = wave * 16 + lo;        // B/D column for this lane

  v8f acc = {};
#pragma unroll
  for (int k0 = 0; k0 < 64; k0 += 4) {
    const int ka = k0 + 2 * hi;
    v2f a, b;
    a.x = A[(size_t)row * 64 + ka];
    a.y = A[(size_t)row * 64 + ka + 1];
    b.x = B[(size_t)ka * 128 + col];
    b.y = B[(size_t)(ka + 1) * 128 + col];
    acc = __builtin_amdgcn_wmma_f32_16x16x4_f32(
        /*neg_a=*/false, a, /*neg_b=*/false, b,
        /*c_mod=*/(short)0, acc, /*reuse_a=*/false, /*reuse_b=*/false);
  }
#pragma unroll
  for (int v = 0; v < 8; ++v) {
    int r = mt * 16 + hi * 8 + v;
    D[(size_t)r * 128 + col] = acc[v];
  }
}

// ---------------------------------------------------------------------------
// Layer 2 node prep: attention dots over feat2; reset emax/denom; zero agg2.
// ---------------------------------------------------------------------------
__global__ void l2_node_prep(const float* __restrict__ feat2,
                             const float* __restrict__ asrc, const float* __restrict__ adst,
                             float* __restrict__ as_, float* __restrict__ ad_,
                             unsigned* __restrict__ emax, float* __restrict__ denom,
                             float* __restrict__ agg2) {
  int n = blockIdx.x * blockDim.x + threadIdx.x;
  if (n >= NNODES) return;
#pragma unroll
  for (int h = 0; h < 4; ++h) {
    float s = 0.f, d = 0.f;
#pragma unroll
    for (int c = 0; c < 32; ++c) {
      float hv = feat2[(size_t)n * 128 + h * 32 + c];
      s += hv * asrc[h * 32 + c];
      d += hv * adst[h * 32 + c];
    }
    as_[n * 4 + h]   = s;
    ad_[n * 4 + h]   = d;
    emax[n * 4 + h]  = 0u;
    denom[n * 4 + h] = 0.0f;
  }
#pragma unroll 4
  for (int f = 0; f < 128; ++f) agg2[(size_t)n * 128 + f] = 0.0f;
}

__global__ void zero_out(float* __restrict__ out) { out[threadIdx.x] = 0.0f; }

// bias + tanh + global mean pool -> out[128]
__global__ void bias_tanh_pool(const float* __restrict__ agg2, const float* __restrict__ b2,
                               float* __restrict__ out) {
  int f = threadIdx.x;                      // 128 threads
  float acc = 0.0f;
  for (int n = blockIdx.x; n < NNODES; n += gridDim.x)
    acc += tanhf(agg2[(size_t)n * 128 + f] + b2[f]);
  atomicAdd(&out[f], acc * (1.0f / (float)NNODES));
}

// ---------------------------------------------------------------------------
extern "C" void kernel_launch(void* const* d_in, const int* in_sizes, int n_in,
                              void* d_out, int out_size, void* d_ws, size_t ws_size,
                              hipStream_t stream) {
  const float*     x    = (const float*)d_in[0];
  const long long* ei   = (const long long*)d_in[1];   // int64 [2,E]
  const float*     W1   = (const float*)d_in[2];
  const float*     aS1  = (const float*)d_in[3];
  const float*     aD1  = (const float*)d_in[4];
  const float*     b1   = (const float*)d_in[5];
  const float*     W2   = (const float*)d_in[6];
  const float*     aS2  = (const float*)d_in[7];
  const float*     aD2  = (const float*)d_in[8];
  const float*     b2   = (const float*)d_in[9];
  float*           out  = (float*)d_out;

  // Workspace layout (floats). Region A (128N) is reused: layer 1 uses it as
  // feat1 (64N) + z1 (64N); after the GEMM consumes z1 it becomes agg2/z2 (128N).
  float* W     = (float*)d_ws;
  float* feat1 = W;                                  // 64N
  float* z1    = W + (size_t)64 * NNODES;            // 64N (agg1 -> tanh -> z1)
  float* feat2 = W + (size_t)128 * NNODES;           // 128N
  float* agg2  = feat1;                              // reuse region A, 128N
  float* aux   = W + (size_t)256 * NNODES;
  float*    as_   = aux;                             // 4N
  float*    ad_   = aux + (size_t)4 * NNODES;        // 4N
  unsigned* emax  = (unsigned*)(aux + (size_t)8 * NNODES); // 4N
  float*    denom = aux + (size_t)12 * NNODES;       // 4N
  (void)ws_size; (void)in_sizes; (void)n_in; (void)out_size;

  const int nodeBlocks = (NNODES + 255) / 256;
  const long long ehTotal = (long long)ETOT * 4;
  const int ehBlocks = (int)((ehTotal + 255) / 256);

  // ---- layer 1 ----
  l1_node_prep<<<nodeBlocks, 256, 0, stream>>>(x, W1, aS1, aD1, feat1, z1, as_, ad_, emax, denom);
  edge_max<<<ehBlocks, 256, 0, stream>>>(ei, as_, ad_, emax);
  edge_sum<<<ehBlocks, 256, 0, stream>>>(ei, as_, ad_, emax, denom);
  edge_scatter<64, 16><<<ETOT, 64, 0, stream>>>(ei, as_, ad_, emax, denom, feat1, z1);
  bias_tanh<64><<<(int)(((long long)NNODES * 64 + 255) / 256), 256, 0, stream>>>(z1, b1);

  // ---- layer 2 ----
  gemm_l2_wmma<<<NNODES / 16, 256, 0, stream>>>(z1, W2, feat2);   // 6250 M-tiles
  l2_node_prep<<<nodeBlocks, 256, 0, stream>>>(feat2, aS2, aD2, as_, ad_, emax, denom, agg2);
  edge_max<<<ehBlocks, 256, 0, stream>>>(ei, as_, ad_, emax);
  edge_sum<<<ehBlocks, 256, 0, stream>>>(ei, as_, ad_, emax, denom);
  edge_scatter<128, 32><<<ETOT, 128, 0, stream>>>(ei, as_, ad_, emax, denom, feat2, agg2);

  // ---- pool ----
  zero_out<<<1, 128, 0, stream>>>(out);
  bias_tanh_pool<<<512, 128, 0, stream>>>(agg2, b2, out);
}